// ScanLikeModel_36395552867093
// MI455X (gfx1250) — compile-verified
//
#include <hip/hip_runtime.h>

// Problem constants (from reference: L=16384, D=4096, fp32).
#define L_ROWS 16384
#define D_COLS 4096
#define CHUNKS 64
#define ROWS_PER_CHUNK 256   // L_ROWS / CHUNKS
#define TILES_PER_CHUNK 16   // ROWS_PER_CHUNK / 16

typedef float v2f __attribute__((ext_vector_type(2)));
typedef float v8f __attribute__((ext_vector_type(8)));

// ---------------------------------------------------------------------------
// Phase 1: per-chunk column sums.  partial[chunk][d] = sum of 256 rows.
// Grid: (CHUNKS, D/1024), block 256, each thread owns one float4 column group.
// ---------------------------------------------------------------------------
__global__ void __launch_bounds__(256)
scan_phase1_chunksum(const float* __restrict__ seq, float* __restrict__ partial) {
  const int chunk   = blockIdx.x;
  const int col4    = blockIdx.y * blockDim.x + threadIdx.x;  // float4 index
  const int stride4 = D_COLS / 4;
  const float4* p = reinterpret_cast<const float4*>(seq) +
                    (size_t)chunk * ROWS_PER_CHUNK * stride4 + col4;
  float4 s = make_float4(0.f, 0.f, 0.f, 0.f);
  for (int r = 0; r < ROWS_PER_CHUNK; ++r) {
    // Speculative-ish prefetch a few rows ahead (clamped inside the chunk).
    const int pr = (r + 24 < ROWS_PER_CHUNK) ? (r + 24) : r;
    __builtin_prefetch(p + (size_t)pr * stride4, 0, 0);
    float4 v = p[(size_t)r * stride4];
    s.x += v.x; s.y += v.y; s.z += v.z; s.w += v.w;
  }
  reinterpret_cast<float4*>(partial)[(size_t)chunk * stride4 + col4] = s;
}

// ---------------------------------------------------------------------------
// Phase 2: in-place exclusive scan of chunk sums per column, seeded by state.
// After this, partial[chunk][d] = state[d] + sum of all chunks < chunk.
// Also emits final_state[d] = state[d] + total column sum.
// Grid: D/256 blocks of 256 threads, one column per thread.
// ---------------------------------------------------------------------------
__global__ void __launch_bounds__(256)
scan_phase2_offsets(const float* __restrict__ state,
                    float* __restrict__ partial,
                    float* __restrict__ final_state) {
  const int d = blockIdx.x * blockDim.x + threadIdx.x;
  float run = state[d];
  for (int c = 0; c < CHUNKS; ++c) {
    const size_t idx = (size_t)c * D_COLS + d;
    const float v = partial[idx];
    partial[idx] = run;   // exclusive offset for this chunk
    run += v;
  }
  final_state[d] = run;
}

// ---------------------------------------------------------------------------
// Phase 3: WMMA tile prefix.  Each wave owns a 256-row x 16-col region.
// Inclusive prefix of a 16x16 tile = T(16x16 lower-triangular ones) * X + C,
// with C = per-column carry broadcast down all 16 rows.  T is split into four
// 16x4 A-slices for V_WMMA_F32_16X16X4_F32 (K=4 fp32), chained on the
// accumulator.  Carry chains across the 16 tiles via a lane shuffle of row 15.
// Grid: (CHUNKS, D/128), block 256 = 8 waves (one 16-col tile per wave).
// ---------------------------------------------------------------------------
__global__ void __launch_bounds__(256)
scan_phase3_wmma(const float* __restrict__ seq,
                 const float* __restrict__ offsets,
                 float* __restrict__ out) {
  const int lane = threadIdx.x & 31;
  const int wave = threadIdx.x >> 5;
  const int hi   = lane >> 4;    // lane half selects K offset (+2) / M offset (+8)
  const int m    = lane & 15;
  const int chunk = blockIdx.x;
  const int col   = blockIdx.y * 128 + wave * 16 + m;

  // A-matrix slices of the lower-triangular ones matrix.
  // 32-bit A 16x4 layout: lane -> M = lane%16, VGPR v -> K = v + 2*(lane>=16).
  v2f a[4];
#pragma unroll
  for (int s = 0; s < 4; ++s) {
#pragma unroll
    for (int v = 0; v < 2; ++v) {
      a[s][v] = (4 * s + v + 2 * hi) <= m ? 1.0f : 0.0f;
    }
  }

  float carry = offsets[(size_t)chunk * D_COLS + col];  // includes state[]

  const size_t rowBase = (size_t)chunk * ROWS_PER_CHUNK;
  for (int t = 0; t < TILES_PER_CHUNK; ++t) {
    const size_t r0 = rowBase + (size_t)t * 16;

    // B-matrix slices: 4x16 fp32, lane -> N = lane%16, VGPR v -> K = v + 2*hi.
    v2f b[4];
#pragma unroll
    for (int s = 0; s < 4; ++s) {
#pragma unroll
      for (int v = 0; v < 2; ++v) {
        b[s][v] = seq[(r0 + 4 * s + v + 2 * hi) * (size_t)D_COLS + col];
      }
    }

    // C = carry broadcast to all 16 rows of the column.
    v8f c;
#pragma unroll
    for (int k = 0; k < 8; ++k) c[k] = carry;

    // D = carry + T*X  (inclusive prefix of the tile, carry included).
#pragma unroll
    for (int s = 0; s < 4; ++s) {
      c = __builtin_amdgcn_wmma_f32_16x16x4_f32(
          /*neg_a=*/false, a[s], /*neg_b=*/false, b[s],
          /*c_mod=*/(short)0, c, /*reuse_a=*/false, /*reuse_b=*/false);
    }

    // C/D layout: VGPR k holds M = k (lanes 0-15) / M = k+8 (lanes 16-31).
#pragma unroll
    for (int k = 0; k < 8; ++k) {
      out[(r0 + (size_t)(k + 8 * hi)) * (size_t)D_COLS + col] = 2.0f * c[k];
    }

    // Next tile's carry = row M=15 of this tile: VGPR 7, lanes 16-31.
    carry = __shfl(c[7], 16 + m, 32);
  }
}

// ---------------------------------------------------------------------------
// Host-side launch.  d_in[0] = state (4096 f32), d_in[1] = sequence (L*D f32).
// d_out = [outputs (L*D) | final_state (D)] f32.  d_ws: 1 MB of chunk sums.
// ---------------------------------------------------------------------------
extern "C" void kernel_launch(void* const* d_in, const int* in_sizes, int n_in,
                              void* d_out, int out_size, void* d_ws, size_t ws_size,
                              hipStream_t stream) {
  (void)in_sizes; (void)n_in; (void)out_size; (void)ws_size;
  const float* state = (const float*)d_in[0];
  const float* seq   = (const float*)d_in[1];
  float* out         = (float*)d_out;
  float* final_state = out + (size_t)L_ROWS * D_COLS;
  float* partial     = (float*)d_ws;  // CHUNKS * D_COLS floats = 1 MB

  dim3 g1(CHUNKS, D_COLS / 1024);
  scan_phase1_chunksum<<<g1, 256, 0, stream>>>(seq, partial);

  scan_phase2_offsets<<<D_COLS / 256, 256, 0, stream>>>(state, partial, final_state);

  dim3 g3(CHUNKS, D_COLS / 128);
  scan_phase3_wmma<<<g3, 256, 0, stream>>>(seq, partial, out);
}